// Embeddings_35373350650155
// MI455X (gfx1250) — compile-verified
//
#include <hip/hip_runtime.h>
#include <cstdint>

#define B_     32
#define S_     4096
#define D_     256      // D_MODEL / TOKEN_LEN
#define DM_    512      // D_MODEL
#define SMAX_  255      // MAX_SEGS
#define VOCAB_ 50257

typedef unsigned int v4u __attribute__((ext_vector_type(4)));
typedef int          v8i __attribute__((ext_vector_type(8)));
typedef int          v4i __attribute__((ext_vector_type(4)));

__device__ __forceinline__ float4 f4_add(float4 a, float4 b) {
  return make_float4(a.x + b.x, a.y + b.y, a.z + b.z, a.w + b.w);
}
__device__ __forceinline__ float4 f4_scale(float4 a, float s) {
  return make_float4(a.x * s, a.y * s, a.z * s, a.w * s);
}
__device__ __forceinline__ float4 f4_submul(float4 a, float4 b, float s) {
  return make_float4((a.x - b.x) * s, (a.y - b.y) * s, (a.z - b.z) * s, (a.w - b.w) * s);
}

// ---------------------------------------------------------------------------
// Main kernel: one wave32 per (b, seg). blockIdx.y = batch row, blockIdx.x*4
// + wave = segment slot. TDM gather-mode DMA pulls the segment's <=16
// embedding rows (1KB each) from the L2-resident table into this wave's LDS
// slice; wave reduces with ds_load_b128. Lane L owns dims [8L, 8L+8).
// ---------------------------------------------------------------------------
__global__ __launch_bounds__(128) void seg_kernel(const int* __restrict__ x,
                                                  const float* __restrict__ table,
                                                  const float* __restrict__ pad_emb,
                                                  const int* __restrict__ nseg,
                                                  const int* __restrict__ starts,
                                                  float* __restrict__ out) {
  __shared__ __align__(16) float smem[4 * 16 * D_];  // 4 waves x 16 rows x 1KB = 64KB

  const int lane = threadIdx.x & 31;
  const int wv   = __builtin_amdgcn_readfirstlane((int)(threadIdx.x >> 5));
  const int b    = blockIdx.y;                       // [0, 32)
  const int s    = blockIdx.x * 4 + wv;              // [0, 256) -> guard
  if (s >= SMAX_) return;

  float* const op = out + ((size_t)b * SMAX_ + s) * DM_;
  const int ns    = nseg[b];

  if (s >= ns) {                                   // unemitted -> pad_emb*sqrt(512)
    const float k    = 22.627416997969522f;        // sqrt(512)
    const float4* pe = (const float4*)(pad_emb + lane * 16);
    float4* o        = (float4*)(op + lane * 16);
#pragma unroll
    for (int i = 0; i < 4; ++i) o[i] = f4_scale(pe[i], k);
    return;
  }

  const int st0 = starts[b * (SMAX_ + 1) + s];
  const int st1 = starts[b * (SMAX_ + 1) + s + 1];
  const int cnt = st1 - st0;                       // >= 1; == 16 for this data

  float4 f0, f1, a0, a1;

  if (cnt <= 16) {
    // ---- TDM gather-mode path -------------------------------------------
    int tok = 0;
    if (lane < cnt) tok = x[(size_t)b * S_ + st0 + lane];
#define RL(i) __builtin_amdgcn_readlane(tok, i)
    v4i g2 = { (RL(0) & 0xFFFF) | (RL(1) << 16),
               (RL(2) & 0xFFFF) | (RL(3) << 16),
               (RL(4) & 0xFFFF) | (RL(5) << 16),
               (RL(6) & 0xFFFF) | (RL(7) << 16) };
    v4i g3 = { (RL(8)  & 0xFFFF) | (RL(9)  << 16),
               (RL(10) & 0xFFFF) | (RL(11) << 16),
               (RL(12) & 0xFFFF) | (RL(13) << 16),
               (RL(14) & 0xFFFF) | (RL(15) << 16) };
#undef RL
    float* const myrows   = smem + wv * (16 * D_);
    const uint32_t ldsadr = (uint32_t)(uintptr_t)myrows;  // generic LDS addr low 32 = DS offset
    const uint64_t ga     = (uint64_t)(uintptr_t)table;
    v4u g0 = { 0x80000001u,                               // count=1 | gather_mode, 16-bit idx
               ldsadr,
               (uint32_t)ga,
               (uint32_t)((ga >> 32) & 0x01FFFFFFu) | 0x80000000u };  // addr[56:32] | type=2
    v8i g1 = { 0x00020000,                       // data_size = 4B
               (int)(D_ << 16),                  // tensor_dim0[15:0] (=256) in bits 63:48
               (int)((VOCAB_ & 0xFFFF) << 16),   // tensor_dim1[15:0] in bits 95:80
               (int)(D_ << 16),                  // tile_dim0 (=256) in bits 127:112
               cnt,                              // tile_dim1 = #valid indices
               D_,                               // tensor_dim0_stride[31:0]
               0, 0 };
    asm volatile("" ::: "memory");               // fence: TDM writes LDS out-of-band
#if __clang_major__ >= 23
    v8i gpad = {0, 0, 0, 0, 0, 0, 0, 0};
    __builtin_amdgcn_tensor_load_to_lds(g0, g1, g2, g3, gpad, 0);
#else
    __builtin_amdgcn_tensor_load_to_lds(g0, g1, g2, g3, 0);
#endif
    __builtin_amdgcn_s_wait_tensorcnt(0);
    asm volatile("" ::: "memory");               // keep ds_loads below the wait

    const float* base = myrows + lane * 8;
    f0 = *(const float4*)(base);                 // row 0 = first token of segment
    f1 = *(const float4*)(base + 4);
    a0 = f0; a1 = f1;
    for (int r = 1; r < cnt; ++r) {
      a0 = f4_add(a0, *(const float4*)(base + r * D_));
      a1 = f4_add(a1, *(const float4*)(base + r * D_ + 4));
    }
  } else {
    // ---- general fallback: direct coalesced VMEM gather ------------------
    const int* tp    = x + (size_t)b * S_ + st0;
    const float* r0p = table + (size_t)tp[0] * D_ + lane * 8;
    f0 = *(const float4*)(r0p);
    f1 = *(const float4*)(r0p + 4);
    a0 = f0; a1 = f1;
    for (int t = 1; t < cnt; ++t) {
      const float* rt = table + (size_t)tp[t] * D_ + lane * 8;
      a0 = f4_add(a0, *(const float4*)(rt));
      a1 = f4_add(a1, *(const float4*)(rt + 4));
    }
  }

  const float scale = 16.0f;                     // sqrt(D_)
  float4 r0v, r1v;
  if (cnt > 1) {
    const float inv = 1.0f / (float)(cnt - 1);
    r0v = f4_submul(a0, f0, inv);
    r1v = f4_submul(a1, f1, inv);
  } else {
    r0v = f0; r1v = f1;
  }
  *(float4*)(op + lane * 8)            = f4_scale(f0, scale);
  *(float4*)(op + lane * 8 + 4)        = f4_scale(f1, scale);
  *(float4*)(op + D_ + lane * 8)       = f4_scale(r0v, scale);
  *(float4*)(op + D_ + lane * 8 + 4)   = f4_scale(r1v, scale);
}

// ---------------------------------------------------------------------------
// Kernel 1: one wave32 per row. Find split positions (token==1) before the
// first pad (token==0). starts[b][0] = 0, starts[b][k] = pos of k-th split.
// Segment s spans [starts[s], starts[s+1]) for s < n_segs.
// ---------------------------------------------------------------------------
__global__ __launch_bounds__(32) void scan_kernel(const int* __restrict__ x,
                                                  int* __restrict__ nseg,
                                                  int* __restrict__ starts) {
  const int b    = blockIdx.x;
  const int lane = threadIdx.x;
  const int* row = x + (size_t)b * S_;
  int* st        = starts + b * (SMAX_ + 1);

  int count = 0;
  bool stop = false;
  for (int chunk = 0; chunk < S_ && !stop; chunk += 32) {
    const int p   = chunk + lane;
    const int tok = row[p];
    unsigned padm = __builtin_amdgcn_ballot_w32(tok == 0);
    unsigned splm = __builtin_amdgcn_ballot_w32(tok == 1);
    if (padm) {
      const unsigned fp = (unsigned)__builtin_ctz(padm);       // first pad lane
      splm = (fp == 0u) ? 0u : (splm & (0xFFFFFFFFu >> (32 - fp)));
      stop = true;
    }
    const unsigned below = (lane == 0) ? 0u : (splm & (0xFFFFFFFFu >> (32 - lane)));
    if (splm & (1u << lane)) {
      const int k = count + __popc(below) + 1;
      if (k <= SMAX_) st[k] = p;
    }
    count += __popc(splm);
  }
  if (lane == 0) {
    nseg[b] = (count > SMAX_) ? SMAX_ : count;
    st[0]   = 0;
  }
}

// ---------------------------------------------------------------------------
extern "C" void kernel_launch(void* const* d_in, const int* in_sizes, int n_in,
                              void* d_out, int out_size, void* d_ws, size_t ws_size,
                              hipStream_t stream) {
  (void)in_sizes; (void)n_in; (void)out_size; (void)ws_size;
  const int*   x       = (const int*)d_in[0];
  const float* table   = (const float*)d_in[1];
  const float* pad_emb = (const float*)d_in[2];
  // d_in[3] = max_segs (== 255, fixed by the reference module)

  int* nseg   = (int*)d_ws;            // [32]
  int* starts = nseg + B_;             // [32][256]

  scan_kernel<<<B_, 32, 0, stream>>>(x, nseg, starts);

  // 2-D grid: y = batch row, x*4 + wave = segment slot (256 slots, 255 used)
  dim3 grid(64, B_, 1);
  seg_kernel<<<grid, 128, 0, stream>>>(x, table, pad_emb, nseg,
                                       starts, (float*)d_out);
}